// Net_15908558864825
// MI455X (gfx1250) — compile-verified
//
#include <hip/hip_runtime.h>
#include <math.h>
#include <stdint.h>

typedef __attribute__((ext_vector_type(2))) float v2f;
typedef __attribute__((ext_vector_type(8))) float v8f;

__device__ __forceinline__ void atomic_add_f32(float* p, float v) {
  (void)__hip_atomic_fetch_add(p, v, __ATOMIC_RELAXED, __HIP_MEMORY_SCOPE_AGENT);
}

// ---------------- zero workspace region ----------------
__global__ void k_zero(float* __restrict__ p, long n) {
  long i = (long)blockIdx.x * blockDim.x + threadIdx.x;
  if (i < n) p[i] = 0.0f;
}

// ---------------- degree (count on dst, fp32 exact up to 2^24) ----------------
__global__ void k_degree(const int* __restrict__ dst, int E, float* __restrict__ deg) {
  int e = blockIdx.x * blockDim.x + threadIdx.x;
  if (e < E) atomic_add_f32(&deg[dst[e]], 1.0f);
}

// ---------------- dinv = rsqrt(deg + 1) in place ----------------
__global__ void k_rsqrt(float* __restrict__ d, int n) {
  int i = blockIdx.x * blockDim.x + threadIdx.x;
  if (i < n) d[i] = rsqrtf(d[i] + 1.0f);
}

// ---------------- GEMM1: h1[N,16] = x[N,512] @ W1[512,16] via V_WMMA_F32_16X16X4_F32 ----
// One wave per 16-row tile. A (16x4 f32): lane m = lane&15, vgpr v holds K = v + 2*(lane>>4).
// B (4x16): vgpr v holds row K = v + 2*(lane>>4), col = lane&15.
// D (16x16 f32): vgpr r holds row r + 8*(lane>>4), col = lane&15.
// W1 (32KB) is staged into LDS with gfx1250 async global->LDS loads (ASYNCcnt path).
__global__ void __launch_bounds__(256) k_gemm1(const float* __restrict__ x,
                                               const float* __restrict__ W1,
                                               float* __restrict__ h1,
                                               int ntiles, int n_nodes) {
  __shared__ float sW1[512 * 16];   // 32 KB of the 320 KB WGP LDS

  // --- async-stage W1 into LDS: each thread copies 32 elements (8192 total / 256 threads)
  {
    int t = threadIdx.x;
#pragma unroll
    for (int i = 0; i < 32; ++i) {
      int e = t + i * 256;
      uint64_t g = (uint64_t)(uintptr_t)(W1 + e);
      uint32_t l = (uint32_t)(uintptr_t)(&sW1[e]);   // low 32 bits = LDS byte address
      asm volatile("global_load_async_to_lds_b32 %0, %1, off"
                   :: "v"(l), "v"(g) : "memory");
    }
    asm volatile("s_wait_asynccnt 0x0" ::: "memory");
  }
  __syncthreads();

  int wave = (int)((blockIdx.x * blockDim.x + threadIdx.x) >> 5);
  int lane = threadIdx.x & 31;
  if (wave >= ntiles) return;                 // wave-uniform: EXEC stays all-ones
  int row0 = wave * 16;
  int m    = lane & 15;
  int kgrp = (lane >> 4) << 1;                // 0 or 2
  int rrow = row0 + m; if (rrow > n_nodes - 1) rrow = n_nodes - 1;  // clamp reads
  const float* xrow = x + (size_t)rrow * 512 + kgrp;

  v8f c = {};
  for (int k0 = 0; k0 < 512; k0 += 4) {
    if ((k0 & 63) == 0) __builtin_prefetch(xrow + k0 + 96, 0, 0);   // global_prefetch_b8
    v2f a = *(const v2f*)(xrow + k0);          // K = k0+kgrp, k0+kgrp+1 (8B aligned)
    v2f b;
    b.x = sW1[(k0 + kgrp) * 16 + m];           // ds_load
    b.y = sW1[(k0 + kgrp + 1) * 16 + m];
    c = __builtin_amdgcn_wmma_f32_16x16x4_f32(false, a, false, b, (short)0, c, false, false);
  }

  int half = lane >> 4;
  if (row0 + 16 <= n_nodes) {                  // uniform fast path: no divergent stores
    float* outp = h1 + (size_t)(row0 + half * 8) * 16 + m;
#pragma unroll
    for (int r = 0; r < 8; ++r) outp[(size_t)r * 16] = c[r];
  } else {
#pragma unroll
    for (int r = 0; r < 8; ++r) {
      int row = row0 + r + half * 8;
      if (row < n_nodes) h1[(size_t)row * 16 + m] = c[r];
    }
  }
}

// ---------------- edge scatter, 16 features, 4 threads per edge ----------------
__global__ void k_scatter16(const int* __restrict__ src, const int* __restrict__ dst, int E,
                            const float* __restrict__ dinv, const float* __restrict__ h,
                            float* __restrict__ agg) {
  long gid = (long)blockIdx.x * blockDim.x + threadIdx.x;
  if (gid >= (long)E * 4) return;
  int e = (int)(gid >> 2);
  int q = ((int)gid & 3) << 2;                 // 0,4,8,12
  int s = src[e], d = dst[e];
  float w = dinv[s] * dinv[d];
  float4 v = *(const float4*)(h + (size_t)s * 16 + q);
  float* a = agg + (size_t)d * 16 + q;
  atomic_add_f32(a + 0, w * v.x);
  atomic_add_f32(a + 1, w * v.y);
  atomic_add_f32(a + 2, w * v.z);
  atomic_add_f32(a + 3, w * v.w);
}

// ---------------- fused: hrelu = relu(agg1 + dinv^2*h1 + b1); h2 = hrelu @ W2 (W2 padded N=16)
__global__ void __launch_bounds__(256) k_finalize1(const float* __restrict__ agg1,
                                                   const float* __restrict__ h1,
                                                   const float* __restrict__ dinv,
                                                   const float* __restrict__ b1,
                                                   const float* __restrict__ W2,
                                                   float* __restrict__ h2,
                                                   int ntiles, int n_nodes) {
  __shared__ float sW2[16 * 16];               // W2 zero-padded to 16x16
  {
    int t = threadIdx.x;                       // blockDim == 256 == 16*16
    int k = t >> 4, n = t & 15;
    sW2[t] = (n < 7) ? W2[(size_t)k * 7 + n] : 0.0f;
  }
  __syncthreads();

  int wave = (int)((blockIdx.x * blockDim.x + threadIdx.x) >> 5);
  int lane = threadIdx.x & 31;
  if (wave >= ntiles) return;                  // wave-uniform
  int row0 = wave * 16;
  int m    = lane & 15;
  int kgrp = (lane >> 4) << 1;
  int rrow = row0 + m; if (rrow > n_nodes - 1) rrow = n_nodes - 1;
  float dv  = dinv[rrow];
  float sii = dv * dv;
  const float* ag = agg1 + (size_t)rrow * 16;
  const float* hh = h1   + (size_t)rrow * 16;

  v8f c = {};
#pragma unroll
  for (int k0 = 0; k0 < 16; k0 += 4) {
    int ka = k0 + kgrp;
    v2f a, b;
    a.x = fmaxf(ag[ka]     + sii * hh[ka]     + b1[ka],     0.0f);
    a.y = fmaxf(ag[ka + 1] + sii * hh[ka + 1] + b1[ka + 1], 0.0f);
    b.x = sW2[ka * 16 + m];
    b.y = sW2[(ka + 1) * 16 + m];
    c = __builtin_amdgcn_wmma_f32_16x16x4_f32(false, a, false, b, (short)0, c, false, false);
  }
  if (m < 7) {
    int half = lane >> 4;
#pragma unroll
    for (int r = 0; r < 8; ++r) {
      int row = row0 + r + half * 8;
      if (row < n_nodes) h2[(size_t)row * 7 + m] = c[r];
    }
  }
}

// ---------------- edge scatter, 7 features, 2 threads per edge ----------------
__global__ void k_scatter7(const int* __restrict__ src, const int* __restrict__ dst, int E,
                           const float* __restrict__ dinv, const float* __restrict__ h,
                           float* __restrict__ agg) {
  long gid = (long)blockIdx.x * blockDim.x + threadIdx.x;
  if (gid >= (long)E * 2) return;
  int e = (int)(gid >> 1);
  int q = ((int)gid & 1) << 2;                 // 0 or 4
  int nf = (q == 0) ? 4 : 3;
  int s = src[e], d = dst[e];
  float w = dinv[s] * dinv[d];
  const float* hv = h + (size_t)s * 7 + q;
  float* a = agg + (size_t)d * 7 + q;
  for (int j = 0; j < nf; ++j) atomic_add_f32(a + j, w * hv[j]);
}

// ---------------- out = log_softmax(agg2 + dinv^2*h2 + b2) ----------------
__global__ void k_finalize2(const float* __restrict__ agg2, const float* __restrict__ h2,
                            const float* __restrict__ dinv, const float* __restrict__ b2,
                            float* __restrict__ out, int n) {
  int i = blockIdx.x * blockDim.x + threadIdx.x;
  if (i >= n) return;
  float dv = dinv[i];
  float sii = dv * dv;
  float v[7];
  float mx = -1e30f;
#pragma unroll
  for (int c = 0; c < 7; ++c) {
    float t = agg2[(size_t)i * 7 + c] + sii * h2[(size_t)i * 7 + c] + b2[c];
    v[c] = t;
    mx = fmaxf(mx, t);
  }
  float sum = 0.0f;
#pragma unroll
  for (int c = 0; c < 7; ++c) sum += __expf(v[c] - mx);
  float lse = mx + __logf(sum);
#pragma unroll
  for (int c = 0; c < 7; ++c) out[(size_t)i * 7 + c] = v[c] - lse;
}

extern "C" void kernel_launch(void* const* d_in, const int* in_sizes, int n_in,
                              void* d_out, int out_size, void* d_ws, size_t ws_size,
                              hipStream_t stream) {
  const float* x  = (const float*)d_in[0];
  const int*   ei = (const int*)d_in[1];
  const float* W1 = (const float*)d_in[2];
  const float* b1 = (const float*)d_in[3];
  const float* W2 = (const float*)d_in[4];
  const float* b2 = (const float*)d_in[5];
  float* out = (float*)d_out;

  int n_nodes = in_sizes[0] / 512;
  int E       = in_sizes[1] / 2;
  const int* src = ei;
  const int* dst = ei + E;

  // workspace layout: [dinv N][agg1 16N][agg2 7N][h1 16N][h2 7N]  (first 24N zeroed each call)
  float* wsf  = (float*)d_ws;
  float* dinv = wsf;
  float* agg1 = dinv + n_nodes;
  float* agg2 = agg1 + (size_t)n_nodes * 16;
  float* h1   = agg2 + (size_t)n_nodes * 7;
  float* h2   = h1   + (size_t)n_nodes * 16;

  const int T = 256;
  long nz = (long)n_nodes * 24;
  k_zero<<<(int)((nz + T - 1) / T), T, 0, stream>>>(wsf, nz);

  k_degree<<<(E + T - 1) / T, T, 0, stream>>>(dst, E, dinv);
  k_rsqrt<<<(n_nodes + T - 1) / T, T, 0, stream>>>(dinv, n_nodes);

  int ntiles = (n_nodes + 15) / 16;
  int gwaves = (ntiles * 32 + T - 1) / T;
  k_gemm1<<<gwaves, T, 0, stream>>>(x, W1, h1, ntiles, n_nodes);

  long s1 = (long)E * 4;
  k_scatter16<<<(int)((s1 + T - 1) / T), T, 0, stream>>>(src, dst, E, dinv, h1, agg1);

  k_finalize1<<<gwaves, T, 0, stream>>>(agg1, h1, dinv, b1, W2, h2, ntiles, n_nodes);

  long s2 = (long)E * 2;
  k_scatter7<<<(int)((s2 + T - 1) / T), T, 0, stream>>>(src, dst, E, dinv, h2, agg2);

  k_finalize2<<<(n_nodes + T - 1) / T, T, 0, stream>>>(agg2, h2, dinv, b2, out, n_nodes);
}